// GAT_45535243273030
// MI455X (gfx1250) — compile-verified
//
#include <hip/hip_runtime.h>
#include <hip/hip_bf16.h>

// ---------------------------------------------------------------------------
// GAT-like network, N=4096 (64x64), nfeat=128, nhid=64, nclass=16, 3 heads.
// All N x N similarity work runs on V_WMMA_F32_16X16X4_F32 (CDNA5 fp32 WMMA).
// e = h1 h1^T is symmetric, so column norms come from one WMMA pass (squares
// reduced per column); the normalized product is a second WMMA pass against
// w = h1 / colnorm (pre-scaled, stored transposed for contiguous B-fragments).
// This revision double-buffers fragment loads so WMMA chains overlap memory.
// ---------------------------------------------------------------------------

typedef float v2f __attribute__((ext_vector_type(2)));
typedef float v8f __attribute__((ext_vector_type(8)));

#define N_PIX  4096
#define NHID   64
#define NCLS   16

__device__ __forceinline__ v8f wmma4(v2f a, v2f b, v8f c) {
  // D(16x16,f32) = A(16x4,f32) * B(4x16,f32) + C
  return __builtin_amdgcn_wmma_f32_16x16x4_f32(false, a, false, b, (short)0, c,
                                               false, false);
}

__device__ __forceinline__ v8f vzero8() {
  v8f z = {0.f, 0.f, 0.f, 0.f, 0.f, 0.f, 0.f, 0.f};
  return z;
}

// Load KS fragments (A or B layout; identical addressing) from a row-major
// matrix with leading dim ld, 16-row tile starting at row0.
template <int KS>
__device__ __forceinline__ void load_frags(v2f (&f)[KS],
                                           const float* __restrict__ base,
                                           int row0, int ld, int r, int hi) {
#pragma unroll
  for (int ks = 0; ks < KS; ++ks)
    f[ks] = *(const v2f*)(base + (size_t)(row0 + r) * ld + 4 * ks + 2 * hi);
}

template <int KS>
__device__ __forceinline__ v8f mmchain(const v2f (&a)[KS], const v2f (&b)[KS]) {
  v8f s = vzero8();
#pragma unroll
  for (int ks = 0; ks < KS; ++ks) s = wmma4(a[ks], b[ks], s);
  return s;
}

__device__ __forceinline__ void acc_sq(v8f& sq, const v8f& s) {
#pragma unroll
  for (int v = 0; v < 8; ++v) sq[v] += s[v] * s[v];
}

// ---------------------------------------------------------------------------
// k1: per-head hidden h = xf @ W^T + b, then row-L2-normalize -> h1buf.
// grid (4096, 3) x 64 threads. x is (128, 4096) row-major; xf[n][f]=x[f*N+n].
// ---------------------------------------------------------------------------
__global__ void k1_hidden(const float* __restrict__ x,
                          const float* __restrict__ W1, const float* __restrict__ W2,
                          const float* __restrict__ W3,
                          const float* __restrict__ b1, const float* __restrict__ b2,
                          const float* __restrict__ b3,
                          float* __restrict__ h1buf) {
  const int n = blockIdx.x, head = blockIdx.y, k = threadIdx.x;
  const float* W = (head == 0) ? W1 : (head == 1 ? W2 : W3);
  const float* b = (head == 0) ? b1 : (head == 1 ? b2 : b3);
  float acc = b[k];
  const float* wr = W + (size_t)k * 128;
#pragma unroll 8
  for (int f = 0; f < 128; ++f) acc += x[(size_t)f * N_PIX + n] * wr[f];
  __shared__ float red[64];
  red[k] = acc * acc;
  __syncthreads();
  for (int s = 32; s > 0; s >>= 1) {
    if (k < s) red[k] += red[k + s];
    __syncthreads();
  }
  const float inv = 1.f / fmaxf(sqrtf(red[0]), 1e-12f);
  h1buf[((size_t)head * N_PIX + n) * NHID + k] = acc * inv;
}

// ---------------------------------------------------------------------------
// k2a: partial column norms^2 of e = h1 h1^T (K=64).
// grid (jt=256, ichunk=4, head=3) x 32. Double-buffered i-tile streaming.
// ---------------------------------------------------------------------------
__global__ void k2a_colnorm(const float* __restrict__ h1buf,
                            float* __restrict__ pcn2) {
  const int jt = blockIdx.x, ic = blockIdx.y, head = blockIdx.z;
  const float* h1 = h1buf + (size_t)head * N_PIX * NHID;
  const int lane = threadIdx.x, r = lane & 15, hi = lane >> 4;
  const int jbase = jt * 16;
  v2f bfr[16];
  load_frags(bfr, h1, jbase, NHID, r, hi);
  const int i0 = ic * 64, iEnd = i0 + 64;
  v2f fa[16], fb[16];
  load_frags(fa, h1, i0 * 16, NHID, r, hi);
  v8f sq = vzero8();
  for (int it = i0; it < iEnd; it += 2) {
    load_frags(fb, h1, (it + 1) * 16, NHID, r, hi);   // prefetch odd tile
    v8f s = mmchain(fa, bfr);
    acc_sq(sq, s);
    const int itn = (it + 2 < iEnd) ? it + 2 : it;
    load_frags(fa, h1, itn * 16, NHID, r, hi);        // prefetch next even
    s = mmchain(fb, bfr);
    acc_sq(sq, s);
  }
  float cs = 0.f;
#pragma unroll
  for (int v = 0; v < 8; ++v) cs += sq[v];
  cs += __shfl_xor(cs, 16, 32);  // combine row halves (same column n)
  if (lane < 16) pcn2[((size_t)ic * 3 + head) * N_PIX + jbase + lane] = cs;
}

// k2b: wT[head][k][j] = h1[head][j][k] / max(sqrt(cn2[j]), eps). grid (4096,3)x64
__global__ void k2b_scale(const float* __restrict__ h1buf,
                          const float* __restrict__ pcn2,
                          float* __restrict__ wT) {
  const int j = blockIdx.x, head = blockIdx.y, k = threadIdx.x;
  float cn2 = 0.f;
#pragma unroll
  for (int c = 0; c < 4; ++c) cn2 += pcn2[((size_t)c * 3 + head) * N_PIX + j];
  const float icn = 1.f / fmaxf(sqrtf(cn2), 1e-12f);
  wT[((size_t)head * NHID + k) * N_PIX + j] =
      h1buf[((size_t)head * N_PIX + j) * NHID + k] * icn;
}

// ---------------------------------------------------------------------------
// k3a: out_partial = (h1 h1^T) @ w over a j-chunk. grid (it=256, jchunk=4,
// head=3) x 32. Per tile: wb loads issued first (overlap with 16 S-WMMAs),
// S(16x16) LDS-bounced to A-layout, 16 WMMAs accumulate S @ w into 16x64.
// ---------------------------------------------------------------------------
__device__ __forceinline__ void k3_tile(const v2f (&a)[16], const v2f (&bf)[16],
                                        const float* __restrict__ wt, int jbase,
                                        int r, int hi, float* sS, v8f (&acc)[4]) {
  // Issue independent w-fragment loads first; they complete during the S chain.
  v2f wb[16];
#pragma unroll
  for (int nt = 0; nt < 4; ++nt)
#pragma unroll
    for (int ks2 = 0; ks2 < 4; ++ks2)
      wb[nt * 4 + ks2] = *(const v2f*)(wt + (size_t)(r + 16 * nt) * N_PIX +
                                       jbase + 4 * ks2 + 2 * hi);
  v8f s = mmchain(a, bf);
  // C-layout (m = v + 8*hi, n = r) -> LDS -> reload as A fragments
#pragma unroll
  for (int v = 0; v < 8; ++v) sS[(v + 8 * hi) * 16 + r] = s[v];
  __syncthreads();
  v2f sa[4];
#pragma unroll
  for (int ks2 = 0; ks2 < 4; ++ks2)
    sa[ks2] = *(const v2f*)(&sS[r * 16 + 4 * ks2 + 2 * hi]);
  __syncthreads();
#pragma unroll
  for (int nt = 0; nt < 4; ++nt)
#pragma unroll
    for (int ks2 = 0; ks2 < 4; ++ks2)
      acc[nt] = wmma4(sa[ks2], wb[nt * 4 + ks2], acc[nt]);
}

__global__ void k3a_attmm(const float* __restrict__ h1buf,
                          const float* __restrict__ wT,
                          float* __restrict__ pacc) {
  __shared__ float sS[256];
  const int it = blockIdx.x, jc = blockIdx.y, head = blockIdx.z;
  const float* h1 = h1buf + (size_t)head * N_PIX * NHID;
  const float* wt = wT + (size_t)head * NHID * N_PIX;
  const int lane = threadIdx.x, r = lane & 15, hi = lane >> 4;
  const int ibase = it * 16;
  v2f a[16];
  load_frags(a, h1, ibase, NHID, r, hi);
  v8f acc[4];
#pragma unroll
  for (int t = 0; t < 4; ++t) acc[t] = vzero8();

  const int j0 = jc * 64, jEnd = j0 + 64;
  v2f bf0[16], bf1[16];
  load_frags(bf0, h1, j0 * 16, NHID, r, hi);
  for (int jt = j0; jt < jEnd; jt += 2) {
    load_frags(bf1, h1, (jt + 1) * 16, NHID, r, hi);   // prefetch odd tile
    k3_tile(a, bf0, wt, jt * 16, r, hi, sS, acc);
    const int jn = (jt + 2 < jEnd) ? jt + 2 : jt;
    load_frags(bf0, h1, jn * 16, NHID, r, hi);         // prefetch next even
    k3_tile(a, bf1, wt, (jt + 1) * 16, r, hi, sS, acc);
  }
  float* p = pacc + (((size_t)jc * 3 + head) * N_PIX + ibase) * NHID;
#pragma unroll
  for (int nt = 0; nt < 4; ++nt)
#pragma unroll
    for (int v = 0; v < 8; ++v)
      p[(size_t)(v + 8 * hi) * NHID + r + 16 * nt] = acc[nt][v];
}

// k3b: sum 4 j-chunk partials, apply ELU, write concat layout hc[N][192].
__global__ void k3b_elu(const float* __restrict__ pacc, float* __restrict__ hcbuf) {
  const int idx = blockIdx.x * 256 + threadIdx.x;  // < 3*4096*64
  const int k = idx & 63, i = (idx >> 6) & 4095, head = idx >> 18;
  float s = 0.f;
#pragma unroll
  for (int c = 0; c < 4; ++c)
    s += pacc[(((size_t)c * 3 + head) * N_PIX + i) * NHID + k];
  hcbuf[(size_t)i * (3 * NHID) + head * NHID + k] = (s > 0.f) ? s : expm1f(s);
}

// ---------------------------------------------------------------------------
// k4: class head linear g = hc @ Wo^T + bo, row-normalize -> h1o[N][16].
// grid 1024 x 64 (4 rows x 16 classes per block).
// ---------------------------------------------------------------------------
__global__ void k4_outlin(const float* __restrict__ hcbuf,
                          const float* __restrict__ Wo,
                          const float* __restrict__ bo,
                          float* __restrict__ h1o) {
  const int t = threadIdx.x, c = t & 15, rr = t >> 4;
  const int n = blockIdx.x * 4 + rr;
  float acc = bo[c];
  const float* hr = hcbuf + (size_t)n * (3 * NHID);
  const float* wr = Wo + (size_t)c * (3 * NHID);
#pragma unroll 8
  for (int f = 0; f < 3 * NHID; ++f) acc += hr[f] * wr[f];
  __shared__ float red[4][16];
  red[rr][c] = acc * acc;
  __syncthreads();
  for (int s = 8; s > 0; s >>= 1) {
    if (c < s) red[rr][c] += red[rr][c + s];
    __syncthreads();
  }
  const float inv = 1.f / fmaxf(sqrtf(red[rr][0]), 1e-12f);
  h1o[(size_t)n * NCLS + c] = acc * inv;
}

// k5a: partial column norms^2 for the class head (K=16). grid (256,4) x 32.
__global__ void k5a_colnorm(const float* __restrict__ h1o, float* __restrict__ pcno2) {
  const int jt = blockIdx.x, ic = blockIdx.y;
  const int lane = threadIdx.x, r = lane & 15, hi = lane >> 4;
  const int jbase = jt * 16;
  v2f bfr[4];
  load_frags(bfr, h1o, jbase, NCLS, r, hi);
  const int i0 = ic * 64, iEnd = i0 + 64;
  v2f fa[4], fb[4];
  load_frags(fa, h1o, i0 * 16, NCLS, r, hi);
  v8f sq = vzero8();
  for (int it = i0; it < iEnd; it += 2) {
    load_frags(fb, h1o, (it + 1) * 16, NCLS, r, hi);
    v8f s = mmchain(fa, bfr);
    acc_sq(sq, s);
    const int itn = (it + 2 < iEnd) ? it + 2 : it;
    load_frags(fa, h1o, itn * 16, NCLS, r, hi);
    s = mmchain(fb, bfr);
    acc_sq(sq, s);
  }
  float cs = 0.f;
#pragma unroll
  for (int v = 0; v < 8; ++v) cs += sq[v];
  cs += __shfl_xor(cs, 16, 32);
  if (lane < 16) pcno2[(size_t)ic * N_PIX + jbase + lane] = cs;
}

// k5b: woT[c][j] = h1o[j][c] / max(sqrt(cn2), eps). grid 4096 x 16.
__global__ void k5b_scale(const float* __restrict__ h1o,
                          const float* __restrict__ pcno2,
                          float* __restrict__ woT) {
  const int j = blockIdx.x, c = threadIdx.x;
  float cn2 = 0.f;
#pragma unroll
  for (int q = 0; q < 4; ++q) cn2 += pcno2[(size_t)q * N_PIX + j];
  const float icn = 1.f / fmaxf(sqrtf(cn2), 1e-12f);
  woT[(size_t)c * N_PIX + j] = h1o[(size_t)j * NCLS + c] * icn;
}

// k6a: final att @ h1o, split over j-chunks. grid (256,4) x 32.
__device__ __forceinline__ void k6_tile(const v2f (&a)[4], const v2f (&bf)[4],
                                        const float* __restrict__ woT, int jbase,
                                        int r, int hi, float* sS, v8f& acc) {
  v2f wb[4];
#pragma unroll
  for (int ks2 = 0; ks2 < 4; ++ks2)
    wb[ks2] = *(const v2f*)(woT + (size_t)r * N_PIX + jbase + 4 * ks2 + 2 * hi);
  v8f s = mmchain(a, bf);
#pragma unroll
  for (int v = 0; v < 8; ++v) sS[(v + 8 * hi) * 16 + r] = s[v];
  __syncthreads();
  v2f sa[4];
#pragma unroll
  for (int ks2 = 0; ks2 < 4; ++ks2)
    sa[ks2] = *(const v2f*)(&sS[r * 16 + 4 * ks2 + 2 * hi]);
  __syncthreads();
#pragma unroll
  for (int ks2 = 0; ks2 < 4; ++ks2) acc = wmma4(sa[ks2], wb[ks2], acc);
}

__global__ void k6a_attmm(const float* __restrict__ h1o,
                          const float* __restrict__ woT,
                          float* __restrict__ pacco) {
  __shared__ float sS[256];
  const int it = blockIdx.x, jc = blockIdx.y;
  const int lane = threadIdx.x, r = lane & 15, hi = lane >> 4;
  const int ibase = it * 16;
  v2f a[4];
  load_frags(a, h1o, ibase, NCLS, r, hi);
  v8f acc = vzero8();
  const int j0 = jc * 64, jEnd = j0 + 64;
  v2f bf0[4], bf1[4];
  load_frags(bf0, h1o, j0 * 16, NCLS, r, hi);
  for (int jt = j0; jt < jEnd; jt += 2) {
    load_frags(bf1, h1o, (jt + 1) * 16, NCLS, r, hi);
    k6_tile(a, bf0, woT, jt * 16, r, hi, sS, acc);
    const int jn = (jt + 2 < jEnd) ? jt + 2 : jt;
    load_frags(bf0, h1o, jn * 16, NCLS, r, hi);
    k6_tile(a, bf1, woT, (jt + 1) * 16, r, hi, sS, acc);
  }
  float* p = pacco + ((size_t)jc * N_PIX + ibase) * NCLS;
#pragma unroll
  for (int v = 0; v < 8; ++v) p[(size_t)(v + 8 * hi) * NCLS + r] = acc[v];
}

// k6b: sum partials, write transposed output out[c*N + i]. grid 256 x 256.
__global__ void k6b_out(const float* __restrict__ pacco, float* __restrict__ out) {
  const int idx = blockIdx.x * 256 + threadIdx.x;  // < 65536
  const int c = idx & 15, i = idx >> 4;
  float s = 0.f;
#pragma unroll
  for (int q = 0; q < 4; ++q) s += pacco[((size_t)q * N_PIX + i) * NCLS + c];
  out[(size_t)c * N_PIX + i] = s;
}

// ---------------------------------------------------------------------------
extern "C" void kernel_launch(void* const* d_in, const int* in_sizes, int n_in,
                              void* d_out, int out_size, void* d_ws, size_t ws_size,
                              hipStream_t stream) {
  (void)in_sizes; (void)n_in; (void)out_size; (void)ws_size;
  const float* x  = (const float*)d_in[0];
  const float* W1 = (const float*)d_in[1];
  const float* b1 = (const float*)d_in[2];
  const float* W2 = (const float*)d_in[3];
  const float* b2 = (const float*)d_in[4];
  const float* W3 = (const float*)d_in[5];
  const float* b3 = (const float*)d_in[6];
  const float* Wo = (const float*)d_in[7];
  const float* bo = (const float*)d_in[8];
  float* out = (float*)d_out;

  // workspace layout (floats); total ~5.96M floats (~23.9 MB)
  float* ws    = (float*)d_ws;
  float* h1buf = ws;                  // 3*4096*64      = 786432
  float* wT    = h1buf + 786432;      // 3*64*4096      = 786432
  float* pcn2  = wT + 786432;         // 4*3*4096       = 49152
  float* pacc  = pcn2 + 49152;        // 4*3*4096*64    = 3145728
  float* hcbuf = pacc + 3145728;      // 4096*192       = 786432
  float* h1o   = hcbuf + 786432;      // 4096*16        = 65536
  float* pcno2 = h1o + 65536;         // 4*4096         = 16384
  float* woT   = pcno2 + 16384;       // 16*4096        = 65536
  float* pacco = woT + 65536;         // 4*4096*16      = 262144

  k1_hidden<<<dim3(N_PIX, 3), 64, 0, stream>>>(x, W1, W2, W3, b1, b2, b3, h1buf);
  k2a_colnorm<<<dim3(256, 4, 3), 32, 0, stream>>>(h1buf, pcn2);
  k2b_scale<<<dim3(N_PIX, 3), 64, 0, stream>>>(h1buf, pcn2, wT);
  k3a_attmm<<<dim3(256, 4, 3), 32, 0, stream>>>(h1buf, wT, pacc);
  k3b_elu<<<3072, 256, 0, stream>>>(pacc, hcbuf);
  k4_outlin<<<1024, 64, 0, stream>>>(hcbuf, Wo, bo, h1o);
  k5a_colnorm<<<dim3(256, 4), 32, 0, stream>>>(h1o, pcno2);
  k5b_scale<<<N_PIX, 16, 0, stream>>>(h1o, pcno2, woT);
  k6a_attmm<<<dim3(256, 4), 32, 0, stream>>>(h1o, woT, pacco);
  k6b_out<<<256, 256, 0, stream>>>(pacco, out);
}